// D2RL_Critic_33844342293317
// MI455X (gfx1250) — compile-verified
//
#include <hip/hip_runtime.h>
#include <math.h>

typedef float v2f __attribute__((ext_vector_type(2)));
typedef float v8f __attribute__((ext_vector_type(8)));

#define N_NODES 100000
#define N_EDGES 1600000
#define NGRAPH  256
#define EPSV    1e-5f

// workspace layout (in floats)
#define OFF_DEG   0
#define OFF_AGG   (N_NODES)              // N*32
#define OFF_H     (33*N_NODES)           // N*16
#define OFF_AGG2  (49*N_NODES)           // N*16
#define OFF_GSUM  (65*N_NODES)           // G*16
#define OFF_CNT   (65*N_NODES + 16*NGRAPH)
#define OFF_STAT  (65*N_NODES + 16*NGRAPH + NGRAPH)   // 32 floats (sum[16], sumsq[16] -> scale/shift)
#define WS_FLOATS (65*N_NODES + 16*NGRAPH + NGRAPH + 32)

__global__ void k_zero(float* __restrict__ p, int n) {
    int i = blockIdx.x * blockDim.x + threadIdx.x;
    if (i < n) p[i] = 0.0f;
}

// edge scatter for conv1: deg count + agg += x[src]  (8 threads/edge, float4 each)
__global__ void k_scatter1(const float* __restrict__ x, const int* __restrict__ ei,
                           float* __restrict__ deg, float* __restrict__ agg) {
    int tid = blockIdx.x * blockDim.x + threadIdx.x;
    int e = tid >> 3, q = tid & 7;
    if (e >= N_EDGES) return;
    int src = ei[e];
    int dst = ei[N_EDGES + e];
    float4 v = ((const float4*)x)[src * 8 + q];
    float* a = agg + dst * 32 + q * 4;
    atomicAdd(a + 0, v.x); atomicAdd(a + 1, v.y);
    atomicAdd(a + 2, v.z); atomicAdd(a + 3, v.w);
    if (q == 0) atomicAdd(deg + dst, 1.0f);
}

// conv1: h = relu((agg/deg)@W1l^T + x@W1r^T + b1), plus BN stat accumulation.
// One wave per 16-node tile; fp32 WMMA 16x16x4, K=32 in 8 chunks x 2 GEMMs.
__global__ void k_conv1(const float* __restrict__ agg, const float* __restrict__ x,
                        const float* __restrict__ deg,
                        const float* __restrict__ W1l, const float* __restrict__ W1r,
                        const float* __restrict__ b1,
                        float* __restrict__ h, float* __restrict__ stat) {
    __shared__ float ssum[16], ssq[16];
    if (threadIdx.x < 16) { ssum[threadIdx.x] = 0.0f; ssq[threadIdx.x] = 0.0f; }
    __syncthreads();

    int lane = threadIdx.x & 31;
    int tile = blockIdx.x * (blockDim.x >> 5) + (threadIdx.x >> 5);
    if (tile < (N_NODES / 16)) {                 // wave-uniform branch (EXEC all 1s inside)
        int m  = lane & 15;
        int kh = (lane >> 4) * 2;                // K sub-offset per half-wave
        int rowA = tile * 16 + m;
        float rd = 1.0f / fmaxf(deg[rowA], 1.0f);
        int n = m;                               // B/C column
        v8f acc = {};
        #pragma unroll
        for (int kk = 0; kk < 32; kk += 4) {
            int k0 = kk + kh;
            v2f a1, b1v, a2, b2v;
            a1[0] = agg[rowA * 32 + k0] * rd;  a1[1] = agg[rowA * 32 + k0 + 1] * rd;
            b1v[0] = W1l[n * 32 + k0];         b1v[1] = W1l[n * 32 + k0 + 1];
            acc = __builtin_amdgcn_wmma_f32_16x16x4_f32(false, a1, false, b1v,
                                                        (short)0, acc, false, false);
            a2[0] = x[rowA * 32 + k0];         a2[1] = x[rowA * 32 + k0 + 1];
            b2v[0] = W1r[n * 32 + k0];         b2v[1] = W1r[n * 32 + k0 + 1];
            acc = __builtin_amdgcn_wmma_f32_16x16x4_f32(false, a2, false, b2v,
                                                        (short)0, acc, false, false);
        }
        float bias = b1[n];
        #pragma unroll
        for (int r = 0; r < 8; r++) {
            int row = tile * 16 + r + 8 * (lane >> 4);
            float val = fmaxf(acc[r] + bias, 0.0f);
            h[row * 16 + n] = val;
            atomicAdd(&ssum[n], val);
            atomicAdd(&ssq[n], val * val);
        }
    }
    __syncthreads();
    if (threadIdx.x < 16) {
        atomicAdd(stat + threadIdx.x,      ssum[threadIdx.x]);
        atomicAdd(stat + 16 + threadIdx.x, ssq[threadIdx.x]);
    }
}

// fold BN stats into per-feature scale/shift (stat[0..15]=scale, stat[16..31]=shift)
__global__ void k_bnfix(const float* __restrict__ g1, const float* __restrict__ be1,
                        float* __restrict__ stat) {
    int f = threadIdx.x;
    if (f < 16) {
        float inv = 1.0f / (float)N_NODES;
        float mean = stat[f] * inv;
        float var  = stat[16 + f] * inv - mean * mean;
        float sc = g1[f] * rsqrtf(var + EPSV);
        float sh = be1[f] - mean * sc;
        stat[f] = sc;
        stat[16 + f] = sh;
    }
}

// edge scatter for conv2 with fused BN on the gathered features (4 threads/edge)
__global__ void k_scatter2(const float* __restrict__ h, const int* __restrict__ ei,
                           const float* __restrict__ stat, float* __restrict__ agg2) {
    int tid = blockIdx.x * blockDim.x + threadIdx.x;
    int e = tid >> 2, q = tid & 3;
    if (e >= N_EDGES) return;
    int src = ei[e];
    int dst = ei[N_EDGES + e];
    float4 v  = ((const float4*)h)[src * 4 + q];
    float4 sc = ((const float4*)stat)[q];
    float4 sh = ((const float4*)(stat + 16))[q];
    float* a = agg2 + dst * 16 + q * 4;
    atomicAdd(a + 0, v.x * sc.x + sh.x);
    atomicAdd(a + 1, v.y * sc.y + sh.y);
    atomicAdd(a + 2, v.z * sc.z + sh.z);
    atomicAdd(a + 3, v.w * sc.w + sh.w);
}

// conv2: h2 = relu((agg2/deg)@W2l^T + bn(h)@W2r^T + b2), fused mean-pool into gsum/cnt.
__global__ void k_conv2(const float* __restrict__ agg2, const float* __restrict__ h,
                        const float* __restrict__ deg, const float* __restrict__ stat,
                        const float* __restrict__ W2l, const float* __restrict__ W2r,
                        const float* __restrict__ b2, const int* __restrict__ batch,
                        float* __restrict__ gsum, float* __restrict__ cnt) {
    int lane = threadIdx.x & 31;
    int tile = blockIdx.x * (blockDim.x >> 5) + (threadIdx.x >> 5);
    if (tile >= (N_NODES / 16)) return;
    int m  = lane & 15;
    int kh = (lane >> 4) * 2;
    int rowA = tile * 16 + m;
    float rd = 1.0f / fmaxf(deg[rowA], 1.0f);
    int n = m;
    v8f acc = {};
    #pragma unroll
    for (int kk = 0; kk < 16; kk += 4) {
        int k0 = kk + kh;
        v2f a1, b1v, a2, b2v;
        a1[0] = agg2[rowA * 16 + k0] * rd;  a1[1] = agg2[rowA * 16 + k0 + 1] * rd;
        b1v[0] = W2l[n * 16 + k0];          b1v[1] = W2l[n * 16 + k0 + 1];
        acc = __builtin_amdgcn_wmma_f32_16x16x4_f32(false, a1, false, b1v,
                                                    (short)0, acc, false, false);
        a2[0] = h[rowA * 16 + k0] * stat[k0] + stat[16 + k0];
        a2[1] = h[rowA * 16 + k0 + 1] * stat[k0 + 1] + stat[16 + k0 + 1];
        b2v[0] = W2r[n * 16 + k0];          b2v[1] = W2r[n * 16 + k0 + 1];
        acc = __builtin_amdgcn_wmma_f32_16x16x4_f32(false, a2, false, b2v,
                                                    (short)0, acc, false, false);
    }
    float bias = b2[n];
    #pragma unroll
    for (int r = 0; r < 8; r++) {
        int row = tile * 16 + r + 8 * (lane >> 4);
        float val = fmaxf(acc[r] + bias, 0.0f);
        int g = batch[row];
        atomicAdd(gsum + g * 16 + n, val);
        if (n == 0) atomicAdd(cnt + g, 1.0f);
    }
}

// D2RL head: one block, thread g owns graph g. LDS-based batch-norm stats.
__device__ inline void bn_block(float* v, int F, const float* __restrict__ gamma,
                                const float* __restrict__ beta,
                                float* ssum, float* ssq) {
    if (threadIdx.x < 32) { ssum[threadIdx.x] = 0.0f; ssq[threadIdx.x] = 0.0f; }
    __syncthreads();
    for (int f = 0; f < F; f++) { atomicAdd(&ssum[f], v[f]); atomicAdd(&ssq[f], v[f] * v[f]); }
    __syncthreads();
    const float inv = 1.0f / (float)NGRAPH;
    for (int f = 0; f < F; f++) {
        float m = ssum[f] * inv;
        float var = ssq[f] * inv - m * m;
        v[f] = (v[f] - m) * rsqrtf(var + EPSV) * gamma[f] + beta[f];
    }
    __syncthreads();
}

__global__ void k_head(const float* __restrict__ gsum, const float* __restrict__ cnt,
                       const float* __restrict__ gn1, const float* __restrict__ bn1,
                       const float* __restrict__ Wl1, const float* __restrict__ bl1,
                       const float* __restrict__ gn2, const float* __restrict__ bn2,
                       const float* __restrict__ Wl2, const float* __restrict__ bl2,
                       const float* __restrict__ gn3, const float* __restrict__ bn3,
                       const float* __restrict__ Wl3, const float* __restrict__ bl3,
                       const float* __restrict__ Wout, const float* __restrict__ bout,
                       float* __restrict__ out) {
    __shared__ float ssum[32], ssq[32];
    int g = threadIdx.x;           // 0..255
    float xe[16], z[32], t[16];
    float c = fmaxf(cnt[g], 1.0f);
    for (int f = 0; f < 16; f++) xe[f] = gsum[g * 16 + f] / c;

    for (int f = 0; f < 16; f++) z[f] = xe[f];
    bn_block(z, 16, gn1, bn1, ssum, ssq);
    for (int i = 0; i < 16; i++) {
        float s = bl1[i];
        for (int j = 0; j < 16; j++) s += z[j] * Wl1[i * 16 + j];
        t[i] = fmaxf(s, 0.0f);
    }
    for (int f = 0; f < 16; f++) { z[f] = t[f]; z[16 + f] = xe[f]; }
    bn_block(z, 32, gn2, bn2, ssum, ssq);
    for (int i = 0; i < 16; i++) {
        float s = bl2[i];
        for (int j = 0; j < 32; j++) s += z[j] * Wl2[i * 32 + j];
        t[i] = fmaxf(s, 0.0f);
    }
    for (int f = 0; f < 16; f++) { z[f] = t[f]; z[16 + f] = xe[f]; }
    bn_block(z, 32, gn3, bn3, ssum, ssq);
    for (int i = 0; i < 16; i++) {
        float s = bl3[i];
        for (int j = 0; j < 32; j++) s += z[j] * Wl3[i * 32 + j];
        t[i] = fmaxf(s, 0.0f);
    }
    float o = bout[0];
    for (int i = 0; i < 16; i++) o += t[i] * Wout[i];
    out[g] = o;
}

extern "C" void kernel_launch(void* const* d_in, const int* in_sizes, int n_in,
                              void* d_out, int out_size, void* d_ws, size_t ws_size,
                              hipStream_t stream) {
    const float* x    = (const float*)d_in[0];
    const int*   ei   = (const int*)d_in[1];
    const int*   batch= (const int*)d_in[2];
    const float* W1l  = (const float*)d_in[3];
    const float* W1r  = (const float*)d_in[4];
    const float* b1   = (const float*)d_in[5];
    const float* g1   = (const float*)d_in[6];
    const float* be1  = (const float*)d_in[7];
    const float* W2l  = (const float*)d_in[8];
    const float* W2r  = (const float*)d_in[9];
    const float* b2   = (const float*)d_in[10];
    const float* gn1  = (const float*)d_in[11];
    const float* bn1  = (const float*)d_in[12];
    const float* Wl1  = (const float*)d_in[13];
    const float* bl1  = (const float*)d_in[14];
    const float* gn2  = (const float*)d_in[15];
    const float* bn2  = (const float*)d_in[16];
    const float* Wl2  = (const float*)d_in[17];
    const float* bl2  = (const float*)d_in[18];
    const float* gn3  = (const float*)d_in[19];
    const float* bn3  = (const float*)d_in[20];
    const float* Wl3  = (const float*)d_in[21];
    const float* bl3  = (const float*)d_in[22];
    const float* Wout = (const float*)d_in[23];
    const float* bout = (const float*)d_in[24];
    float* out = (float*)d_out;

    float* ws   = (float*)d_ws;
    float* deg  = ws + OFF_DEG;
    float* agg  = ws + OFF_AGG;
    float* h    = ws + OFF_H;
    float* agg2 = ws + OFF_AGG2;
    float* gsum = ws + OFF_GSUM;
    float* cnt  = ws + OFF_CNT;
    float* stat = ws + OFF_STAT;

    // 1) zero accumulators
    {
        int n = WS_FLOATS;
        k_zero<<<(n + 255) / 256, 256, 0, stream>>>(ws, n);
    }
    // 2) conv1 edge gather/scatter (+degree)
    {
        long long thr = (long long)N_EDGES * 8;
        k_scatter1<<<(int)((thr + 255) / 256), 256, 0, stream>>>(x, ei, deg, agg);
    }
    // 3) conv1 WMMA + BN stats
    {
        int tiles = N_NODES / 16;                 // 6250
        int blocks = (tiles + 7) / 8;             // 8 waves/block
        k_conv1<<<blocks, 256, 0, stream>>>(agg, x, deg, W1l, W1r, b1, h, stat);
    }
    // 4) fold BN
    k_bnfix<<<1, 16, 0, stream>>>(g1, be1, stat);
    // 5) conv2 edge gather/scatter with fused BN
    {
        long long thr = (long long)N_EDGES * 4;
        k_scatter2<<<(int)((thr + 255) / 256), 256, 0, stream>>>(h, ei, stat, agg2);
    }
    // 6) conv2 WMMA + fused graph mean-pool accumulation
    {
        int tiles = N_NODES / 16;
        int blocks = (tiles + 7) / 8;
        k_conv2<<<blocks, 256, 0, stream>>>(agg2, h, deg, stat, W2l, W2r, b2, batch, gsum, cnt);
    }
    // 7) D2RL head
    k_head<<<1, NGRAPH, 0, stream>>>(gsum, cnt, gn1, bn1, Wl1, bl1, gn2, bn2, Wl2, bl2,
                                     gn3, bn3, Wl3, bl3, Wout, bout, out);
}